// RGCNModel_87454124081943
// MI455X (gfx1250) — compile-verified
//
#include <hip/hip_runtime.h>
#include <math.h>

typedef __attribute__((ext_vector_type(2))) float v2f;
typedef __attribute__((ext_vector_type(8))) float v8f;

constexpr int NNODES = 50000;
constexpr int EMB    = 128;
constexpr int HID    = 64;
constexpr int NRELS  = 16;
constexpr int NCLS   = 16;
constexpr int NEDGES = 800000;

// ---------------------------------------------------------------------------
// Dense transform: for r in [0,NRELS): Y[n, r, :] = X[n,:] @ W[r]
//                  slot r == NRELS   : Hout[n, :] = X[n,:] @ root + bias
// One wave32 computes one 16x16 output tile via V_WMMA_F32_16X16X4_F32.
// A-matrix (16x4 f32): lane L holds M = L&15; VGPR pair = K {0,1} (lanes 0-15)
// or K {2,3} (lanes 16-31).  B-matrix (4x16 f32): mirrored, N = L&15.
// C/D (16x16 f32): VGPR v -> row (L<16 ? v : v+8), col = L&15.
// ---------------------------------------------------------------------------
template <int K, int OUTW, bool RELU_A>
__global__ void rgcn_gemm_kernel(const float* __restrict__ X,     // [NNODES, K]
                                 const float* __restrict__ W,     // [NRELS, K, OUTW]
                                 const float* __restrict__ root,  // [K, OUTW]
                                 const float* __restrict__ bias,  // [OUTW]
                                 float* __restrict__ Y,           // [NNODES, NRELS, OUTW]
                                 float* __restrict__ Hout)        // [NNODES, OUTW]
{
    constexpr int MTILES = NNODES / 16;   // 3125, exact
    constexpr int NTILES = OUTW / 16;
    const int wave = (int)((blockIdx.x * blockDim.x + threadIdx.x) >> 5);
    const int lane = (int)(threadIdx.x & 31);
    const int total = MTILES * NTILES * (NRELS + 1);
    if (wave >= total) return;                 // wave-uniform: EXEC stays all-1

    int t = wave;
    const int mtile = t % MTILES; t /= MTILES;
    const int ntile = t % NTILES; t /= NTILES;
    const int rel   = t;                       // 0..NRELS (NRELS == root slot)

    const int hf   = lane >> 4;                // half-wave select (K pairs)
    const int l16  = lane & 15;
    const int m    = mtile * 16 + l16;         // A row owned by this lane
    const int ncol = ntile * 16 + l16;         // B/D column owned by this lane

    const float* Wp   = (rel < NRELS) ? (W + (size_t)rel * K * OUTW) : root;
    const float* Arow = X + (size_t)m * K + 2 * hf;
    const float* Bcol = Wp + (size_t)(2 * hf) * OUTW + ncol;

    v8f acc = {};
#pragma unroll
    for (int k = 0; k < K; k += 4) {
        v2f a = *(const v2f*)(Arow + k);       // 8B aligned: k%4==0, 2*hf even
        if (RELU_A) { a.x = fmaxf(a.x, 0.0f); a.y = fmaxf(a.y, 0.0f); }
        v2f b;
        b.x = Bcol[(size_t)k * OUTW];
        b.y = Bcol[(size_t)(k + 1) * OUTW];
        acc = __builtin_amdgcn_wmma_f32_16x16x4_f32(
            /*neg_a=*/false, a, /*neg_b=*/false, b,
            /*c_mod=*/(short)0, acc, /*reuse_a=*/false, /*reuse_b=*/false);
    }

    const int rowbase = mtile * 16 + hf * 8;
    if (rel < NRELS) {
        float* Yp = Y + (size_t)rowbase * (NRELS * OUTW) + (size_t)rel * OUTW + ncol;
#pragma unroll
        for (int v = 0; v < 8; ++v)
            Yp[(size_t)v * (NRELS * OUTW)] = acc[v];
    } else {
        const float bv = bias[ncol];
        float* Hp = Hout + (size_t)rowbase * OUTW + ncol;
#pragma unroll
        for (int v = 0; v < 8; ++v)
            Hp[(size_t)v * OUTW] = acc[v] + bv;
    }
}

// ---------------------------------------------------------------------------
// Edge bookkeeping + scatter kernels
// ---------------------------------------------------------------------------
__global__ void count_kernel(const int* __restrict__ dst, const int* __restrict__ et,
                             float* __restrict__ counts)
{
    int e = blockIdx.x * blockDim.x + threadIdx.x;
    if (e < NEDGES) atomicAdd(&counts[dst[e] * NRELS + et[e]], 1.0f);
}

__global__ void invert_kernel(float* __restrict__ counts)
{
    int i = blockIdx.x * blockDim.x + threadIdx.x;
    if (i < NNODES * NRELS) counts[i] = 1.0f / fmaxf(counts[i], 1.0f);
}

// Layer 1 scatter: 16 threads per edge, each moves 4 contiguous f32 (OUTW=64).
__global__ void scatter1_kernel(const int* __restrict__ src, const int* __restrict__ dst,
                                const int* __restrict__ et, const float* __restrict__ inv,
                                const float* __restrict__ Y1, float* __restrict__ H)
{
    int t  = blockIdx.x * blockDim.x + threadIdx.x;
    int e  = t >> 4;
    int o4 = (t & 15) << 2;
    if (e >= NEDGES) return;
    const int s = src[e], d = dst[e], r = et[e];
    const float scale = inv[d * NRELS + r];
    const float4 v = *(const float4*)(Y1 + (size_t)s * (NRELS * HID) + r * HID + o4);
    float* hp = H + (size_t)d * HID + o4;
    atomicAdd(hp + 0, v.x * scale);
    atomicAdd(hp + 1, v.y * scale);
    atomicAdd(hp + 2, v.z * scale);
    atomicAdd(hp + 3, v.w * scale);
}

// Layer 2 scatter: 4 threads per edge, each moves 4 contiguous f32 (OUTW=16).
__global__ void scatter2_kernel(const int* __restrict__ src, const int* __restrict__ dst,
                                const int* __restrict__ et, const float* __restrict__ inv,
                                const float* __restrict__ Y2, float* __restrict__ out)
{
    int t  = blockIdx.x * blockDim.x + threadIdx.x;
    int e  = t >> 2;
    int o4 = (t & 3) << 2;
    if (e >= NEDGES) return;
    const int s = src[e], d = dst[e], r = et[e];
    const float scale = inv[d * NRELS + r];
    const float4 v = *(const float4*)(Y2 + (size_t)s * (NRELS * NCLS) + r * NCLS + o4);
    float* op = out + (size_t)d * NCLS + o4;
    atomicAdd(op + 0, v.x * scale);
    atomicAdd(op + 1, v.y * scale);
    atomicAdd(op + 2, v.z * scale);
    atomicAdd(op + 3, v.w * scale);
}

__global__ void sigmoid_kernel(float* __restrict__ out)
{
    int i = blockIdx.x * blockDim.x + threadIdx.x;
    if (i < NNODES * NCLS) {
        float x = out[i];
        out[i] = 1.0f / (1.0f + __expf(-x));
    }
}

// ---------------------------------------------------------------------------
extern "C" void kernel_launch(void* const* d_in, const int* in_sizes, int n_in,
                              void* d_out, int out_size, void* d_ws, size_t ws_size,
                              hipStream_t stream)
{
    const float* emb   = (const float*)d_in[0];  // [50000,128]
    const float* W1    = (const float*)d_in[1];  // [16,128,64]
    const float* root1 = (const float*)d_in[2];  // [128,64]
    const float* b1    = (const float*)d_in[3];  // [64]
    const float* W2    = (const float*)d_in[4];  // [16,64,16]
    const float* root2 = (const float*)d_in[5];  // [64,16]
    const float* b2    = (const float*)d_in[6];  // [16]
    const int*   eidx  = (const int*)d_in[7];    // [2,800000]
    const int*   etyp  = (const int*)d_in[8];    // [800000]
    const int* src = eidx;
    const int* dst = eidx + NEDGES;
    float* out = (float*)d_out;                  // [50000,16] f32

    // Workspace layout (floats): counts | H | Y2 | Y1   (~272 MB total)
    float* ws     = (float*)d_ws;
    float* counts = ws;                                   //   800,000
    float* H      = counts + (size_t)NNODES * NRELS;      // 3,200,000
    float* Y2     = H + (size_t)NNODES * HID;             // 12,800,000
    float* Y1     = Y2 + (size_t)NNODES * NRELS * NCLS;   // 51,200,000

    hipMemsetAsync(counts, 0, (size_t)NNODES * NRELS * sizeof(float), stream);

    count_kernel<<<(NEDGES + 255) / 256, 256, 0, stream>>>(dst, etyp, counts);
    invert_kernel<<<(NNODES * NRELS + 255) / 256, 256, 0, stream>>>(counts);

    // Layer 1 transform: 3125 M-tiles x 4 N-tiles x 17 slots = 212500 waves
    {
        const long long waves = 3125LL * 4 * (NRELS + 1);
        const int blocks = (int)((waves * 32 + 255) / 256);
        rgcn_gemm_kernel<EMB, HID, false><<<blocks, 256, 0, stream>>>(
            emb, W1, root1, b1, Y1, H);
    }
    scatter1_kernel<<<(NEDGES * 16 + 255) / 256, 256, 0, stream>>>(
        src, dst, etyp, counts, Y1, H);

    // Layer 2 transform (ReLU fused into A loads): 3125 x 1 x 17 = 53125 waves
    {
        const long long waves = 3125LL * 1 * (NRELS + 1);
        const int blocks = (int)((waves * 32 + 255) / 256);
        rgcn_gemm_kernel<HID, NCLS, true><<<blocks, 256, 0, stream>>>(
            H, W2, root2, b2, Y2, out);
    }
    scatter2_kernel<<<(NEDGES * 4 + 255) / 256, 256, 0, stream>>>(
        src, dst, etyp, counts, Y2, out);

    sigmoid_kernel<<<(NNODES * NCLS + 255) / 256, 256, 0, stream>>>(out);
}